// SnapshotGNN_352187319213
// MI455X (gfx1250) — compile-verified
//
#include <hip/hip_runtime.h>

#define T_STEPS 4
#define N_NODES 50000
#define E_EDGES 1600000
#define CH      128
#define GATE3   384
#define OUT_CH  16
#define VSTRIDE 132   // row stride in floats: 528B -> 16B aligned, bank = 4*row+k (conflict-free)

typedef __attribute__((ext_vector_type(2))) float v2f;
typedef __attribute__((ext_vector_type(8))) float v8f;

// ---------------------------------------------------------------------------
// Zero the per-timestep z accumulators (T*CH = 512 floats)
// ---------------------------------------------------------------------------
__global__ void zero_zsum_kernel(float* __restrict__ z) {
    z[threadIdx.x] = 0.0f;
}

// ---------------------------------------------------------------------------
// aggX = x  (self-loop contribution), deg = 1  (self-loop count)
// ---------------------------------------------------------------------------
__global__ void init_agg_kernel(const float* __restrict__ x,
                                float* __restrict__ aggX,
                                float* __restrict__ deg) {
    const int total4 = N_NODES * (CH / 4);
    for (int i = blockIdx.x * blockDim.x + threadIdx.x; i < total4;
         i += gridDim.x * blockDim.x) {
        ((float4*)aggX)[i] = ((const float4*)x)[i];
        if (i < N_NODES) deg[i] = 1.0f;
    }
}

// ---------------------------------------------------------------------------
// Edge scatter: one wave per edge; each lane atomically adds 4 floats.
// aggX (25.6 MB) is L2-resident so these f32 atomics resolve in L2.
// ---------------------------------------------------------------------------
__global__ __launch_bounds__(256) void scatter_kernel(
    const float* __restrict__ x, const int* __restrict__ src,
    const int* __restrict__ dst, float* __restrict__ aggX,
    float* __restrict__ deg) {
    const int lane  = threadIdx.x & 31;
    const int wib   = threadIdx.x >> 5;
    const int waves = (gridDim.x * blockDim.x) >> 5;
    for (int e = blockIdx.x * (blockDim.x >> 5) + wib; e < E_EDGES; e += waves) {
        const int s = src[e];
        const int d = dst[e];
        const float4 v = *(const float4*)(x + (size_t)s * CH + lane * 4);
        float* o = aggX + (size_t)d * CH + lane * 4;
        unsafeAtomicAdd(o + 0, v.x);
        unsafeAtomicAdd(o + 1, v.y);
        unsafeAtomicAdd(o + 2, v.z);
        unsafeAtomicAdd(o + 3, v.w);
        if (lane == 0) unsafeAtomicAdd(deg + d, 1.0f);
    }
}

// ---------------------------------------------------------------------------
// Fused: y = relu((aggX/deg) @ W^T + b); z_sum += column-sum(y) over nodes.
// 8 waves/block; wave w owns output columns [16w,16w+16). Full f32 WMMA.
// Per-row 1/deg scaling is applied to the WMMA accumulator (post-GEMM),
// so the LDS staging is a pure vectorized copy.
// ---------------------------------------------------------------------------
__global__ __launch_bounds__(256) void gemm_mean_kernel(
    const float* __restrict__ aggX, const float* __restrict__ deg,
    const float* __restrict__ Wg, const float* __restrict__ bg,
    float* __restrict__ z_sum) {
    __shared__ __align__(16) float vtile[16 * VSTRIDE];
    __shared__ float rdegs[16];

    const int tid   = threadIdx.x;
    const int lane  = tid & 31;
    const int wv    = tid >> 5;            // 0..7
    const int m     = lane & 15;           // A row / D column within j-tile
    const int kh    = (lane >> 4) << 1;    // 0 or 2 : K sub-pair for this half-wave
    const int j     = (wv << 4) + m;       // output channel 0..127
    const int rbase = (lane >> 4) << 3;    // D rows: rbase..rbase+7

    // Preload this lane's slice of B = W^T (j fixed per lane for whole kernel)
    v2f bw[32];
#pragma unroll
    for (int kk = 0; kk < 32; ++kk) {
        const float* wp = Wg + (size_t)j * CH + (kk << 2) + kh;
        bw[kk] = v2f{wp[0], wp[1]};
    }
    const float bj = bg[j];
    float accum = 0.0f;

    // staging copy: thread i moves 8 consecutive floats (2 x float4), same row
    const int crow = (tid * 8) >> 7;       // row this thread writes
    const int ccol = (tid * 8) & 127;      // starting column

    const int ntiles = N_NODES / 16;       // 3125
    for (int tile = blockIdx.x; tile < ntiles; tile += gridDim.x) {
        const int node0 = tile << 4;
        __syncthreads();
        if (tid < 16) rdegs[tid] = 1.0f / fmaxf(deg[node0 + tid], 1.0f);
        {
            const float4* gsrc =
                (const float4*)(aggX + (size_t)node0 * CH) + tid * 2;
            const float4 a0 = gsrc[0];
            const float4 a1 = gsrc[1];
            *(float4*)&vtile[crow * VSTRIDE + ccol]     = a0;
            *(float4*)&vtile[crow * VSTRIDE + ccol + 4] = a1;
        }
        __syncthreads();

        v8f acc = {};
#pragma unroll
        for (int kk = 0; kk < 32; ++kk) {
            const v2f a = *(const v2f*)(&vtile[m * VSTRIDE + (kk << 2) + kh]);
            acc = __builtin_amdgcn_wmma_f32_16x16x4_f32(
                false, a, false, bw[kk], (short)0, acc, false, false);
        }
        // per-row 1/deg scale + bias + relu + reduce over this lane's 8 rows
#pragma unroll
        for (int r = 0; r < 8; ++r)
            accum += fmaxf(acc[r] * rdegs[rbase + r] + bj, 0.0f);
    }
    // lanes L and L+16 hold disjoint node-rows of the same column j
    accum += __shfl_xor(accum, 16, 32);
    if (lane < 16) unsafeAtomicAdd(&z_sum[j], accum);
}

// ---------------------------------------------------------------------------
// 4 GRU steps + classifier head. Single 128-thread block.
// ---------------------------------------------------------------------------
__global__ __launch_bounds__(128) void gru_head_kernel(
    const float* __restrict__ z_sum,
    const float* __restrict__ W_ih, const float* __restrict__ W_hh,
    const float* __restrict__ b_ih, const float* __restrict__ b_hh,
    const float* __restrict__ W_cls, const float* __restrict__ b_cls,
    float* __restrict__ out) {
    __shared__ float h[CH], z[CH], gi[GATE3], gh[GATE3];
    const int tid = threadIdx.x;
    h[tid] = 0.0f;
    __syncthreads();

    for (int t = 0; t < T_STEPS; ++t) {
        z[tid] = z_sum[t * CH + tid] * (1.0f / (float)N_NODES);
        __syncthreads();
#pragma unroll
        for (int g = 0; g < 3; ++g) {
            const int row = tid + (g << 7);
            float si = b_ih[row];
            float sh = b_hh[row];
            const float* wi = W_ih + (size_t)row * CH;
            const float* wh = W_hh + (size_t)row * CH;
            for (int k = 0; k < CH; ++k) {
                si += wi[k] * z[k];
                sh += wh[k] * h[k];
            }
            gi[row] = si;
            gh[row] = sh;
        }
        __syncthreads();
        const float r  = 1.0f / (1.0f + expf(-(gi[tid] + gh[tid])));
        const float zg = 1.0f / (1.0f + expf(-(gi[CH + tid] + gh[CH + tid])));
        const float ng = tanhf(gi[2 * CH + tid] + r * gh[2 * CH + tid]);
        const float hn = (1.0f - zg) * ng + zg * h[tid];
        __syncthreads();
        h[tid] = hn;
        __syncthreads();
    }

    if (tid < OUT_CH) {
        float s = b_cls[tid];
        const float* wc = W_cls + (size_t)tid * CH;
        for (int k = 0; k < CH; ++k) s += wc[k] * h[k];
        out[tid] = s;
    }
}

// ---------------------------------------------------------------------------
extern "C" void kernel_launch(void* const* d_in, const int* in_sizes, int n_in,
                              void* d_out, int out_size, void* d_ws, size_t ws_size,
                              hipStream_t stream) {
    const float* xs    = (const float*)d_in[0];
    const int*   edges = (const int*)d_in[1];
    const float* Wg    = (const float*)d_in[2];
    const float* bg    = (const float*)d_in[3];
    const float* Wih   = (const float*)d_in[4];
    const float* Whh   = (const float*)d_in[5];
    const float* bih   = (const float*)d_in[6];
    const float* bhh   = (const float*)d_in[7];
    const float* Wcls  = (const float*)d_in[8];
    const float* bcls  = (const float*)d_in[9];
    float* out = (float*)d_out;

    char* ws = (char*)d_ws;
    float* aggX = (float*)(ws);
    float* deg  = (float*)(ws + (size_t)N_NODES * CH * 4);
    float* zsum = (float*)(ws + (size_t)N_NODES * CH * 4 + (size_t)N_NODES * 4);

    zero_zsum_kernel<<<1, T_STEPS * CH, 0, stream>>>(zsum);

    for (int t = 0; t < T_STEPS; ++t) {
        const float* x_t = xs + (size_t)t * N_NODES * CH;
        const int* src = edges + (size_t)t * 2 * E_EDGES;
        const int* dst = src + E_EDGES;
        init_agg_kernel<<<4096, 256, 0, stream>>>(x_t, aggX, deg);
        scatter_kernel<<<25000, 256, 0, stream>>>(x_t, src, dst, aggX, deg);
        gemm_mean_kernel<<<1024, 256, 0, stream>>>(aggX, deg, Wg, bg,
                                                   zsum + t * CH);
    }
    gru_head_kernel<<<1, 128, 0, stream>>>(zsum, Wih, Whh, bih, bhh,
                                           Wcls, bcls, out);
}